// GNNModel_57277683859533
// MI455X (gfx1250) — compile-verified
//
#include <hip/hip_runtime.h>

// ---------------------------------------------------------------------------
// GNN last-timestep pipeline for MI455X (gfx1250, wave32).
// Only t = S-1 contributes to the output, so all GEMMs are [2048 x K] @ [K x 128].
// fp32 WMMA (V_WMMA_F32_16X16X4_F32), operands staged in LDS via CDNA5
// async global->LDS copies (global_load_async_to_lds_b128, ASYNCcnt).
// ---------------------------------------------------------------------------

typedef __attribute__((ext_vector_type(2))) float v2f;
typedef __attribute__((ext_vector_type(8))) float v8f;

#define KC       64                 // K-chunk staged per iteration
#define ASTRIDE  68                 // lds_a row stride (floats): banks 4r+k+2g -> conflict-free b64
#define WSTRIDE  144                // lds_w row stride (floats): banks 16k+32g+r -> conflict-free b32

__device__ __forceinline__ float lrelu(float v) {
    return v > 0.0f ? v : 0.01f * v;
}

// CDNA5 async copy: 16 bytes global -> LDS, tracked by ASYNCcnt.
__device__ __forceinline__ void async_copy_b128(void* lds_dst, const void* gsrc) {
    unsigned lds_off = (unsigned)(unsigned long long)lds_dst; // low 32 bits = LDS byte addr
    unsigned long long ga = (unsigned long long)gsrc;
    asm volatile("global_load_async_to_lds_b128 %0, %1, off"
                 :: "v"(lds_off), "v"(ga) : "memory");
}

__device__ __forceinline__ void async_wait_all() {
    asm volatile("s_wait_asynccnt 0x0" ::: "memory");
}

// Out = act(A @ W [+ A2 @ W2] + bias), A:[M,K] row-major, W:[K,N=128] row-major.
// One block = 8 waves = one 16-row M-slab x all 8 N-tiles. One wave per 16x16 tile.
//
// WMMA f32 16x16x4 fragment layout (ISA 7.12.2, 32-bit):
//   A 16x4 : lane<16 -> row m=lane, K=0,1 ; lane>=16 -> same rows, K=2,3
//   B 4x16 : transpose-symmetric: lane l holds col n=l%16, K-pair 2*(l/16)
//   C 16x16: VGPR i -> row i (lanes 0-15), row i+8 (lanes 16-31), col = lane%16
__global__ void __launch_bounds__(256)
gemm16_wmma_f32(const float* __restrict__ A, const float* __restrict__ W,
                const float* __restrict__ A2, const float* __restrict__ W2,
                const float* __restrict__ bias, float* __restrict__ Out,
                int K, int N, int fuse2, int act)
{
    __shared__ float lds_a[16 * ASTRIDE];      //  ~4.3 KB
    __shared__ float lds_w[KC * WSTRIDE];      // ~36.9 KB

    const int tid  = threadIdx.x;
    const int wave = tid >> 5;                 // 0..7 -> N-tile
    const int lane = tid & 31;
    const int g    = lane >> 4;                // lane group: selects K-pair
    const int r    = lane & 15;                // A row / B,C column within tile
    const int m0   = blockIdx.x << 4;
    const int n0   = wave << 4;

    v8f acc = {};

    for (int p = 0; p <= fuse2; ++p) {
        const float* Ap = p ? A2 : A;
        const float* Wp = p ? W2 : W;

        for (int kc = 0; kc < K; kc += KC) {
            // ---- stage W chunk [KC x N] (row stride padded to WSTRIDE) ----
            for (int idx = tid; idx < KC * (128 / 4); idx += 256) {
                int row = idx >> 5, c4 = (idx & 31) << 2;
                async_copy_b128(&lds_w[row * WSTRIDE + c4],
                                Wp + (size_t)(kc + row) * N + c4);
            }
            // ---- stage A slab [16 x KC] (row stride padded to ASTRIDE) ----
            {
                int row = tid >> 4, c4 = (tid & 15) << 2;   // 256 float4 = 1/thread
                async_copy_b128(&lds_a[row * ASTRIDE + c4],
                                Ap + (size_t)(m0 + row) * K + kc + c4);
            }
            async_wait_all();
            __syncthreads();

            // ---- 16 WMMAs over the staged chunk ----
            for (int k = 0; k < KC; k += 4) {
                v2f a = *(const v2f*)&lds_a[r * ASTRIDE + k + 2 * g];
                v2f b;
                b.x = lds_w[(k + 2 * g) * WSTRIDE + n0 + r];
                b.y = lds_w[(k + 2 * g + 1) * WSTRIDE + n0 + r];
                acc = __builtin_amdgcn_wmma_f32_16x16x4_f32(
                    false, a, false, b, (short)0, acc, false, false);
            }
            __syncthreads();   // LDS reused next chunk/pass
        }
    }

    const float bv = bias[n0 + r];
#pragma unroll
    for (int i = 0; i < 8; ++i) {
        const int m = m0 + i + 8 * g;
        float v = acc[i] + bv;
        if (act) v = lrelu(v);
        Out[(size_t)m * N + n0 + r] = v;
    }
}

__global__ void __launch_bounds__(256)
zero_f32(float* __restrict__ p, int n)
{
    int i = blockIdx.x * blockDim.x + threadIdx.x;
    if (i < n) p[i] = 0.0f;
}

// agg[tgt[e]] += h[src[e]] (32 lanes per edge), cnt[tgt[e]] += 1
__global__ void __launch_bounds__(256)
scatter_accum(const float* __restrict__ h, const int* __restrict__ src,
              const int* __restrict__ tgt, float* __restrict__ agg,
              float* __restrict__ cnt, int E, int H)
{
    int t = blockIdx.x * blockDim.x + threadIdx.x;
    int e = t >> 5;
    int lane = t & 31;
    if (e >= E) return;
    const int s = src[e];
    const int d = tgt[e];
    const float* hs = h + (size_t)s * H;
    float* ad = agg + (size_t)d * H;
    for (int f = lane; f < H; f += 32)
        atomicAdd(&ad[f], hs[f]);
    if (lane == 0) atomicAdd(&cnt[d], 1.0f);
}

__global__ void __launch_bounds__(256)
normalize_rows(float* __restrict__ agg, const float* __restrict__ cnt, int total, int H)
{
    int i = blockIdx.x * blockDim.x + threadIdx.x;
    if (i >= total) return;
    agg[i] /= fmaxf(cnt[i / H], 1.0f);
}

// out[r] = b + sum_f X[r,f] * w[f]
__global__ void __launch_bounds__(256)
matvec_out(const float* __restrict__ X, const float* __restrict__ w,
           const float* __restrict__ b, float* __restrict__ out, int M, int H)
{
    int r = blockIdx.x * blockDim.x + threadIdx.x;
    if (r >= M) return;
    const float* x = X + (size_t)r * H;
    float s = b[0];
    for (int f = 0; f < H; ++f) s = fmaf(x[f], w[f], s);
    out[r] = s;
}

extern "C" void kernel_launch(void* const* d_in, const int* in_sizes, int n_in,
                              void* d_out, int out_size, void* d_ws, size_t ws_size,
                              hipStream_t stream)
{
    (void)in_sizes; (void)n_in; (void)out_size; (void)ws_size;

    const float* x     = (const float*)d_in[0];
    const int*   eidx  = (const int*)d_in[1];   // edge_indexs, per harness int cast
    const float* W1    = (const float*)d_in[3];
    const float* b1    = (const float*)d_in[4];
    const float* W2    = (const float*)d_in[5];
    const float* b2    = (const float*)d_in[6];
    const float* Wrel  = (const float*)d_in[7];
    const float* brel  = (const float*)d_in[8];
    const float* Wroot = (const float*)d_in[9];
    const float* W3    = (const float*)d_in[10];
    const float* b3    = (const float*)d_in[11];
    const float* W4    = (const float*)d_in[12];
    const float* b4    = (const float*)d_in[13];

    const int S = 32, N = 2048, F = 64, H = 128, E = 32768;
    const int t = S - 1;

    // Only timestep t=31 reaches the output.
    const float* x_t = x + (size_t)t * N * F;                  // [2048,64]
    const int* src = eidx + (size_t)t * E;                     // edges[0][t]
    const int* tgt = eidx + (size_t)S * E + (size_t)t * E;     // edges[1][t]

    float* ws  = (float*)d_ws;
    float* h1  = ws;                  // 2048*128
    float* h2  = ws + 262144;         // 2048*128
    float* agg = ws + 524288;         // 2048*128
    float* cnt = ws + 786432;         // 2048 (contiguous after agg for one zero pass)
    float* gnn = ws + 788480;         // 2048*128
    float* o3  = ws + 1050624;        // 2048*128

    const int gblk = N / 16;          // 128 blocks, 8 waves each

    // h1 = lrelu(x_t @ W1 + b1)                         K = 64
    gemm16_wmma_f32<<<gblk, 256, 0, stream>>>(x_t, W1, nullptr, nullptr, b1, h1,
                                              F, H, 0, 1);
    // h2 = lrelu(h1 @ W2 + b2)                          K = 128
    gemm16_wmma_f32<<<gblk, 256, 0, stream>>>(h1, W2, nullptr, nullptr, b2, h2,
                                              H, H, 0, 1);

    // agg/cnt = 0, then scatter-mean over edges of t=31
    {
        int nz = N * H + N;
        zero_f32<<<(nz + 255) / 256, 256, 0, stream>>>(agg, nz);
    }
    scatter_accum<<<(E * 32) / 256, 256, 0, stream>>>(h2, src, tgt, agg, cnt, E, H);
    normalize_rows<<<(N * H + 255) / 256, 256, 0, stream>>>(agg, cnt, N * H, H);

    // gnn = lrelu(agg @ Wrel + h2 @ Wroot + brel)       fused dual GEMM
    gemm16_wmma_f32<<<gblk, 256, 0, stream>>>(agg, Wrel, h2, Wroot, brel, gnn,
                                              H, H, 1, 1);
    // o3 = lrelu(gnn @ W3 + b3)
    gemm16_wmma_f32<<<gblk, 256, 0, stream>>>(gnn, W3, nullptr, nullptr, b3, o3,
                                              H, H, 0, 1);
    // out = o3 @ W4 + b4   -> [2048]
    matvec_out<<<(N + 255) / 256, 256, 0, stream>>>(o3, W4, b4, (float*)d_out, N, H);
}